// History_34488587386982
// MI455X (gfx1250) — compile-verified
//
#include <hip/hip_runtime.h>
#include <hip/hip_bf16.h>

// History.pull on MI455X (gfx1250): pure bandwidth problem (0 FLOPs,
// ~550 MB of traffic -> ~24 us at 23.3 TB/s HBM). Strategy:
//   * one wave32 per row (128 f32 = 32 lanes x float4) -> b128 vmem,
//     wave-uniform cached/uncached branch (zero divergence)
//   * node id made SGPR via readfirstlane -> scalar binary searches over the
//     L2-resident sorted id tables (global_idx: emb row; inter_id: in-batch)
//   * non-temporal loads/stores for the single-touch streams (x, emb, out)
//   * global_prefetch_b8 ahead of the streaming front

typedef __attribute__((ext_vector_type(4))) float v4f;

#define ROW_F32   128          // DIM
#define WAVES_PER_BLOCK 8      // 256 threads = 8 wave32

__global__ __launch_bounds__(256) void history_pull_kernel(
    const float* __restrict__ x,
    const float* __restrict__ emb,
    const int* __restrict__ inter_id,
    const int* __restrict__ layer_id,
    const int* __restrict__ global_idx,
    const unsigned char* __restrict__ cached_nodes,   // jnp bool -> 1 byte
    float* __restrict__ out,
    int M, int n_inter, int num_cache)
{
    const int lane = threadIdx.x & 31;
    const int wave = threadIdx.x >> 5;
    const int row  = blockIdx.x * WAVES_PER_BLOCK + wave;
    if (row >= M) return;

    // Wave-uniform node id, forced into an SGPR so the lookup chain below
    // becomes scalar loads + scalar branches (SMEM path, KMcnt).
    const int id = __builtin_amdgcn_readfirstlane(layer_id[row]);

    // Keep the streaming x front warm in L2 (emits global_prefetch_b8).
    if (lane == 0 && row + 96 < M) {
        __builtin_prefetch(x + (size_t)(row + 96) * ROW_F32, 0, 1);
    }

    // Default source: this row of x.
    const v4f* src = (const v4f*)(x + (size_t)row * ROW_F32);

    if (cached_nodes[id]) {
        // pos lookup: index of id in sorted-unique global_idx (L2 resident,
        // 19 scalar probes). Reference invariant: cached => id in global_idx.
        int lo = 0, hi = num_cache - 1, p = -1;
        while (lo <= hi) {
            const int mid = (lo + hi) >> 1;
            const int v = global_idx[mid];
            if (v == id) { p = mid; break; }
            if (v < id) lo = mid + 1; else hi = mid - 1;
        }
        if (p >= 0) {
            // in_batch test: id present in sorted-unique inter_id.
            int l2 = 0, h2 = n_inter - 1;
            bool inb = false;
            while (l2 <= h2) {
                const int mid = (l2 + h2) >> 1;
                const int v = inter_id[mid];
                if (v == id) { inb = true; break; }
                if (v < id) l2 = mid + 1; else h2 = mid - 1;
            }
            if (inb) src = (const v4f*)(emb + (size_t)p * ROW_F32);
        }
    }

    // One b128 NT load + one b128 NT store per lane = the whole 512 B row
    // per wave. x/emb/out are each touched exactly once -> non-temporal.
    const v4f val = __builtin_nontemporal_load(src + lane);
    __builtin_nontemporal_store(val, (v4f*)(out + (size_t)row * ROW_F32) + lane);
}

extern "C" void kernel_launch(void* const* d_in, const int* in_sizes, int n_in,
                              void* d_out, int out_size, void* d_ws, size_t ws_size,
                              hipStream_t stream) {
    (void)n_in; (void)out_size; (void)d_ws; (void)ws_size;

    const float*         x            = (const float*)d_in[0];
    const float*         emb          = (const float*)d_in[1];
    const int*           inter_id     = (const int*)d_in[2];
    const int*           layer_id     = (const int*)d_in[3];
    const int*           global_idx   = (const int*)d_in[4];
    const unsigned char* cached_nodes = (const unsigned char*)d_in[5];
    float*               out          = (float*)d_out;

    const int n_inter   = in_sizes[2];
    const int M         = in_sizes[3];
    const int num_cache = in_sizes[4];

    const int blocks = (M + WAVES_PER_BLOCK - 1) / WAVES_PER_BLOCK;
    history_pull_kernel<<<blocks, 256, 0, stream>>>(
        x, emb, inter_id, layer_id, global_idx, cached_nodes, out,
        M, n_inter, num_cache);
}